// GraphDependencyParser_66030827209210
// MI455X (gfx1250) — compile-verified
//
#include <hip/hip_runtime.h>
#include <hip/hip_bf16.h>

typedef __attribute__((ext_vector_type(2))) float v2f;
typedef __attribute__((ext_vector_type(8))) float v8f;

// ---------------------------------------------------------------------------
// Problem constants (match reference)
// ---------------------------------------------------------------------------
static constexpr int kB = 32;
static constexpr int kS = 128;
static constexpr int kT = 2000;
static constexpr int kD = 768;
static constexpr int kH = 256;
static constexpr int kR = 50;
static constexpr int kBS = kB * kS;   // 4096 tokens

// ---------------------------------------------------------------------------
// 1) Ragged audio segment mean: aligned[bs, 0:768]
//    One block per token; threads stride over the 768 dims (coalesced).
// ---------------------------------------------------------------------------
__global__ void align_audio_kernel(const float* __restrict__ audio,
                                   const float* __restrict__ st,
                                   const float* __restrict__ et,
                                   float* __restrict__ aligned) {
    int bs = blockIdx.x;
    int b  = bs >> 7;                    // / kS
    int si = (int)(st[bs] * 50.0f);      // trunc == floor for >= 0
    int ei = (int)(et[bs] * 50.0f);
    bool valid = (ei > si) && (ei <= kT);   // block-uniform (scalar branch)
    int sc = min(max(si, 0), kT);
    int ec = min(max(ei, 0), kT);
    int cnt = max(ec - sc, 1);
    float inv = valid ? (1.0f / (float)cnt) : 0.0f;
    const float* base = audio + (long)b * kT * kD;
    for (int d = threadIdx.x; d < kD; d += blockDim.x) {
        float sum = 0.0f;
        if (valid) {
            for (int t = sc; t < ec; ++t) sum += base[(long)t * kD + d];
        }
        aligned[(long)bs * kD + d] = sum * inv;
    }
}

// ---------------------------------------------------------------------------
// 2) Feature assembly: X0[bs, 0:100]=word emb, [100:150]=pos emb, [250:256]=0
//    (cols 150:250 filled later by the audio-projection GEMM)
// ---------------------------------------------------------------------------
__global__ void assemble_features_kernel(const int* __restrict__ words,
                                         const int* __restrict__ pos,
                                         const float* __restrict__ wordW,
                                         const float* __restrict__ posW,
                                         float* __restrict__ X0) {
    int bs = blockIdx.x;
    int t  = threadIdx.x;
    float* x = X0 + (long)bs * 256;
    if (t < 100) {
        x[t] = wordW[(long)words[bs] * 100 + t];
    } else if (t < 150) {
        x[t] = posW[(long)pos[bs] * 50 + (t - 100)];
    } else if (t >= 250) {
        x[t] = 0.0f;   // zero pad -> K=256 tiles are safe, branch-free
    }
}

// ---------------------------------------------------------------------------
// 2b) Weight padding kernels (run once per launch; ~0.8 MB total traffic).
//     Padding the ragged dims lets the GEMM K-loop run with NO guards.
// ---------------------------------------------------------------------------
__global__ void pad_mlp1_kernel(const float* __restrict__ w1,   // [256,250]
                                float* __restrict__ out) {      // [256,256]
    int n = blockIdx.x, k = threadIdx.x;
    out[n * 256 + k] = (k < 250) ? w1[n * 250 + k] : 0.0f;
}

__global__ void pad_audio_w_kernel(const float* __restrict__ aw, // [100,768]
                                   float* __restrict__ out) {    // [128,768]
    int n = blockIdx.x;
    for (int k = threadIdx.x; k < kD; k += blockDim.x)
        out[n * kD + k] = (n < 100) ? aw[n * kD + k] : 0.0f;
}

__global__ void pad_rel_kernel(const float* __restrict__ wr,   // [50,512]
                               float* __restrict__ wrh,        // [64,256]
                               float* __restrict__ wrd) {      // [64,256]
    int n = blockIdx.x, k = threadIdx.x;   // 64 blocks x 256 threads
    bool v = (n < kR);
    wrh[n * 256 + k] = v ? wr[n * 512 + k]       : 0.0f;
    wrd[n * 256 + k] = v ? wr[n * 512 + 256 + k] : 0.0f;
}

// ---------------------------------------------------------------------------
// 3) f32 WMMA GEMM:  C[m,n] = sum_k A[m,k] * W[n,k]  (+bias, relu)
//    One wave per 32x32 tile = 2x2 WMMA subtiles, 4 accumulators
//    (1 global_load_b64 per wmma), with an explicit double-buffered
//    K-pipeline over 32-wide chunks: each 16-WMMA group consumes fragments
//    whose loads were issued one 16-WMMA group earlier, so s_wait_loadcnt is
//    satisfied by compute time instead of stalling.
//    Requirements: M % 32 == 0, Npad % 32 == 0, K % 32 == 0 (all padded).
//    NOTE: the pipeline prefetch over-reads <= 64B past the last K chunk;
//    operands other than w2 and wa live in the workspace, which reserves a tail pad.
//    Fragment layouts per ISA 7.12.2:
//      A 16x4 : lane&15 -> M, (lane>>4)*2 + vgpr -> K
//      B 4x16 : lane&15 -> N, same K split
//      C 16x16: row = m0 + g + 8*(lane>>4), col = n0 + (lane&15)
// ---------------------------------------------------------------------------
__global__ void __launch_bounds__(256)
wmma_gemm_nt_kernel(const float* __restrict__ A, const float* __restrict__ W,
                    const float* __restrict__ bias, float* __restrict__ C,
                    int M, int N, int Npad, int K, int lda, int ldw, int ldc,
                    long sA, long sW, long sC, int relu) {
    int batch = blockIdx.y;
    A += (long)batch * sA;
    W += (long)batch * sW;
    C += (long)batch * sC;

    int tiles_n = Npad >> 5;                         // 32-wide N tiles
    int wave = blockIdx.x * (blockDim.x >> 5) + (threadIdx.x >> 5);
    if (wave >= (M >> 5) * tiles_n) return;          // wave-uniform exit

    int m0 = (wave / tiles_n) << 5;
    int n0 = (wave % tiles_n) << 5;

    int lane = threadIdx.x & 31;
    int half = lane >> 4;          // K sub-block select
    int lm   = lane & 15;          // M row (A) / N col (B)

    const float* Arow0 = A + (long)(m0 + lm) * lda + 2 * half;
    const float* Arow1 = Arow0 + (long)16 * lda;
    const float* Wrow0 = W + (long)(n0 + lm) * ldw + 2 * half;
    const float* Wrow1 = Wrow0 + (long)16 * ldw;

    v8f acc00 = {}, acc01 = {}, acc10 = {}, acc11 = {};

    // Ping-pong fragment buffers (held in registers under full unroll).
    v2f AaA[4], AaB[4], AbA[4], AbB[4];
    v2f BaA[4], BaB[4], BbA[4], BbB[4];

    auto loadA = [&](int koff) {
#pragma unroll
        for (int u = 0; u < 4; ++u) {
            AaA[u] = *(const v2f*)(Arow0 + koff + 4 * u);
            AaB[u] = *(const v2f*)(Arow1 + koff + 4 * u);
            AbA[u] = *(const v2f*)(Wrow0 + koff + 4 * u);
            AbB[u] = *(const v2f*)(Wrow1 + koff + 4 * u);
        }
    };
    auto loadB = [&](int koff) {
#pragma unroll
        for (int u = 0; u < 4; ++u) {
            BaA[u] = *(const v2f*)(Arow0 + koff + 4 * u);
            BaB[u] = *(const v2f*)(Arow1 + koff + 4 * u);
            BbA[u] = *(const v2f*)(Wrow0 + koff + 4 * u);
            BbB[u] = *(const v2f*)(Wrow1 + koff + 4 * u);
        }
    };
    auto computeA = [&]() {
#pragma unroll
        for (int u = 0; u < 4; ++u) {
            acc00 = __builtin_amdgcn_wmma_f32_16x16x4_f32(
                        false, AaA[u], false, AbA[u], (short)0, acc00, false, false);
            acc01 = __builtin_amdgcn_wmma_f32_16x16x4_f32(
                        false, AaA[u], false, AbB[u], (short)0, acc01, false, false);
            acc10 = __builtin_amdgcn_wmma_f32_16x16x4_f32(
                        false, AaB[u], false, AbA[u], (short)0, acc10, false, false);
            acc11 = __builtin_amdgcn_wmma_f32_16x16x4_f32(
                        false, AaB[u], false, AbB[u], (short)0, acc11, false, false);
        }
    };
    auto computeB = [&]() {
#pragma unroll
        for (int u = 0; u < 4; ++u) {
            acc00 = __builtin_amdgcn_wmma_f32_16x16x4_f32(
                        false, BaA[u], false, BbA[u], (short)0, acc00, false, false);
            acc01 = __builtin_amdgcn_wmma_f32_16x16x4_f32(
                        false, BaA[u], false, BbB[u], (short)0, acc01, false, false);
            acc10 = __builtin_amdgcn_wmma_f32_16x16x4_f32(
                        false, BaB[u], false, BbA[u], (short)0, acc10, false, false);
            acc11 = __builtin_amdgcn_wmma_f32_16x16x4_f32(
                        false, BaB[u], false, BbB[u], (short)0, acc11, false, false);
        }
    };

    loadA(0);                               // prologue
    for (int k0 = 0; k0 < K; k0 += 32) {
        loadB(k0 + 16);                     // prefetch second half of chunk
        computeA();                         // 16 wmmas on first half
        loadA(k0 + 32);                     // prefetch next chunk (tail over-read ok)
        computeB();                         // 16 wmmas on second half
    }

    // Epilogue: boundary handling only here (bn guards vs true N).
    int bn0 = n0 + lm;
    int bn1 = bn0 + 16;
    bool v0 = bn0 < N, v1 = bn1 < N;
    float bv0 = (bias && v0) ? bias[bn0] : 0.0f;
    float bv1 = (bias && v1) ? bias[bn1] : 0.0f;
#pragma unroll
    for (int g = 0; g < 8; ++g) {
        int row0 = m0 + g + 8 * half;
        int row1 = row0 + 16;
        if (v0) {
            float a = acc00[g] + bv0;
            float c = acc10[g] + bv0;
            if (relu) { a = a > 0.0f ? a : 0.0f; c = c > 0.0f ? c : 0.0f; }
            C[(long)row0 * ldc + bn0] = a;
            C[(long)row1 * ldc + bn0] = c;
        }
        if (v1) {
            float bq = acc01[g] + bv1;
            float d  = acc11[g] + bv1;
            if (relu) { bq = bq > 0.0f ? bq : 0.0f; d = d > 0.0f ? d : 0.0f; }
            C[(long)row0 * ldc + bn1] = bq;
            C[(long)row1 * ldc + bn1] = d;
        }
    }
}

// ---------------------------------------------------------------------------
// 4) rel_scores[b,i,j,r] = rel_h[b,i,r] + rel_d[b,j,r]   (rel_b folded into
//    rel_h bias). Block per (b,i): rel_h row in LDS, 6400 coalesced stores.
// ---------------------------------------------------------------------------
__global__ void rel_broadcast_kernel(const float* __restrict__ relh,
                                     const float* __restrict__ reld,
                                     float* __restrict__ out) {
    __shared__ float hv[64];
    int bs = blockIdx.x;            // b*kS + i
    int b  = bs >> 7;
    if (threadIdx.x < kR) hv[threadIdx.x] = relh[(long)bs * kR + threadIdx.x];
    __syncthreads();
    const float* dB = reld + (long)b * kS * kR;     // [j, r] contiguous
    float*       o  = out  + (long)bs * kS * kR;
    for (int idx = threadIdx.x; idx < kS * kR; idx += blockDim.x) {
        int r = idx % kR;
        o[idx] = hv[r] + dB[idx];
    }
}

// ---------------------------------------------------------------------------
// Host launcher
// ---------------------------------------------------------------------------
extern "C" void kernel_launch(void* const* d_in, const int* in_sizes, int n_in,
                              void* d_out, int out_size, void* d_ws, size_t ws_size,
                              hipStream_t stream) {
    (void)in_sizes; (void)n_in; (void)out_size; (void)ws_size;

    const int*   words   = (const int*)  d_in[0];
    const int*   pos     = (const int*)  d_in[1];
    const float* audio   = (const float*)d_in[2];
    const float* st      = (const float*)d_in[3];
    const float* et      = (const float*)d_in[4];
    const float* wordW   = (const float*)d_in[5];
    const float* posW    = (const float*)d_in[6];
    const float* audio_w = (const float*)d_in[7];
    const float* audio_b = (const float*)d_in[8];
    const float* w1      = (const float*)d_in[9];
    const float* b1      = (const float*)d_in[10];
    const float* w2      = (const float*)d_in[11];
    const float* b2      = (const float*)d_in[12];
    const float* wa      = (const float*)d_in[13];
    const float* ba      = (const float*)d_in[14];
    const float* wr      = (const float*)d_in[15];   // [50, 512]
    const float* br      = (const float*)d_in[16];

    // Workspace layout (floats); tail pad covers pipeline over-read.
    float* ws      = (float*)d_ws;
    float* aligned = ws;                              // 4096*768
    float* X0      = aligned + (size_t)kBS * kD;      // 4096*256
    float* X1      = X0      + (size_t)kBS * kH;
    float* X2      = X1      + (size_t)kBS * kH;
    float* head    = X2      + (size_t)kBS * kH;
    float* relh    = head    + (size_t)kBS * kH;      // 4096*50
    float* reld    = relh    + (size_t)kBS * kR;
    float* w1p     = reld    + (size_t)kBS * kR;      // [256,256]
    float* awp     = w1p     + (size_t)256 * 256;     // [128,768]
    float* wrhp    = awp     + (size_t)128 * kD;      // [64,256]
    float* wrdp    = wrhp    + (size_t)64 * 256;      // [64,256]
    // (64 floats after wrdp reserved as over-read pad)

    float* arc = (float*)d_out;                       // [32,128,128]
    float* rel = arc + (size_t)kB * kS * kS;          // [32,128,128,50]

    align_audio_kernel<<<kBS, 256, 0, stream>>>(audio, st, et, aligned);
    assemble_features_kernel<<<kBS, 256, 0, stream>>>(words, pos, wordW, posW, X0);
    pad_mlp1_kernel<<<256, 256, 0, stream>>>(w1, w1p);
    pad_audio_w_kernel<<<128, 256, 0, stream>>>(audio_w, awp);
    pad_rel_kernel<<<64, 256, 0, stream>>>(wr, wrhp, wrdp);

    auto gemm = [&](const float* A, const float* Wm, const float* bias, float* C,
                    int M, int N, int Npad, int K, int lda, int ldw, int ldc,
                    long sA, long sW, long sC, int nb, int relu) {
        int tiles  = (M / 32) * (Npad / 32);
        int blocks = (tiles + 7) / 8;                 // 8 waves per 256-thread block
        wmma_gemm_nt_kernel<<<dim3(blocks, nb), 256, 0, stream>>>(
            A, Wm, bias, C, M, N, Npad, K, lda, ldw, ldc, sA, sW, sC, relu);
    };

    // audio projection -> X0 cols [150,250)   (W padded 100->128 rows)
    gemm(aligned, awp, audio_b, X0 + 150, kBS, 100, 128, kD, kD, kD, kH, 0, 0, 0, 1, 0);
    // mlp1 (K padded 250->256 on both operands), relu
    gemm(X0, w1p, b1, X1, kBS, kH, kH, kH, kH, kH, kH, 0, 0, 0, 1, 1);
    // mlp2, relu
    gemm(X1, w2, b2, X2, kBS, kH, kH, kH, kH, kH, kH, 0, 0, 0, 1, 1);
    // arc head projection
    gemm(X2, wa, ba, head, kBS, kH, kH, kH, kH, kH, kH, 0, 0, 0, 1, 0);
    // rel_h = head @ rel_w[:, :256]^T + rel_b   (W padded 50->64 rows)
    gemm(head, wrhp, br, relh, kBS, kR, 64, kH, kH, kH, kR, 0, 0, 0, 1, 0);
    // rel_d = dep  @ rel_w[:, 256:]^T
    gemm(X2, wrdp, nullptr, reld, kBS, kR, 64, kH, kH, kH, kR, 0, 0, 0, 1, 0);
    // arc_scores[b] = head[b] (128x256) @ dep[b]^T  -> batched over 32
    gemm(head, X2, nullptr, arc, kS, kS, kS, kH, kH, kH, kS,
         (long)kS * kH, (long)kS * kH, (long)kS * kS, kB, 0);

    rel_broadcast_kernel<<<kBS, 256, 0, stream>>>(relh, reld, rel);
}